// MoEMLP_59167469470471
// MI455X (gfx1250) — compile-verified
//
#include <hip/hip_runtime.h>
#include <hip/hip_bf16.h>

// Problem sizes (fixed by the reference)
#define H    1280
#define E    896
#define NEXP 64
#define TOPK 6
#define FFN  1792
#define BT   2048

// Padded LDS strides: (X+4) % 64 == 4 -> 16 rows hit 16 distinct banks
#define HLD  (H + 4)
#define ELD  (E + 4)

typedef __attribute__((ext_vector_type(2))) float v2f;
typedef __attribute__((ext_vector_type(4))) float v4f;
typedef __attribute__((ext_vector_type(8))) float v8f;

// D = A(16x4) * B(4x16) + C(16x16), fp32 in / fp32 out (V_WMMA_F32_16X16X4_F32)
__device__ __forceinline__ v8f wmma4(v2f a, v2f b, v8f c) {
  return __builtin_amdgcn_wmma_f32_16x16x4_f32(
      /*neg_a=*/false, a, /*neg_b=*/false, b,
      /*c_mod=*/(short)0, c, /*reuse_a=*/false, /*reuse_b=*/false);
}

// silu without the precise-division ladder: v_exp_f32 + v_rcp_f32 + mul
__device__ __forceinline__ float silu_f(float v) {
  return v * __builtin_amdgcn_rcpf(1.0f + __expf(-v));
}

__global__ void zero_out_kernel(float* __restrict__ p, int n) {
  int i = blockIdx.x * blockDim.x + threadIdx.x;
  if (i < n) p[i] = 0.0f;
}

__global__ void zero_counts_kernel(int* __restrict__ c) {
  c[threadIdx.x] = 0;   // <<<1, NEXP>>>
}

// ---------------------------------------------------------------------------
// Router: logits -> top-6 -> softmax over the top-6 (== renormalized top-k of
// softmax). Produces (a) dense [BT, NEXP] routing matrix (zero outside top-6)
// and (b) per-expert token lists for the sparse routed pass.
// One block (64 threads) per token.
// ---------------------------------------------------------------------------
__global__ void moe_router(const float* __restrict__ x,
                           const float* __restrict__ wr,
                           float* __restrict__ routing,
                           int* __restrict__ counts,
                           int* __restrict__ lists) {
  __shared__ float xs[H];
  __shared__ float logits[NEXP];
  const int b = blockIdx.x;
  for (int i = threadIdx.x; i < H; i += 64) xs[i] = x[(size_t)b * H + i];
  __syncthreads();

  const int e = threadIdx.x;                 // blockDim.x == 64
  const float* w = wr + (size_t)e * H;       // w_router row e, contiguous
  float acc = 0.0f;
#pragma unroll 4
  for (int h = 0; h < H; ++h) acc = __builtin_fmaf(xs[h], w[h], acc);
  logits[e] = acc;
  routing[(size_t)b * NEXP + e] = 0.0f;      // zero-fill; top-6 written below
  __syncthreads();

  if (threadIdx.x == 0) {
    float val[TOPK]; int idx[TOPK];
    unsigned long long used = 0ull;
    for (int k = 0; k < TOPK; ++k) {
      float best = -3.4e38f; int bi = 0;
      for (int n = 0; n < NEXP; ++n)
        if (!((used >> n) & 1ull) && logits[n] > best) { best = logits[n]; bi = n; }
      used |= (1ull << bi); val[k] = best; idx[k] = bi;
    }
    float mx = val[0], sum = 0.0f, p[TOPK];
    for (int k = 0; k < TOPK; ++k) { p[k] = __expf(val[k] - mx); sum += p[k]; }
    float inv = __builtin_amdgcn_rcpf(sum);
    for (int k = 0; k < TOPK; ++k) {
      routing[(size_t)b * NEXP + idx[k]] = p[k] * inv;
      int slot = atomicAdd(&counts[idx[k]], 1);
      lists[idx[k] * BT + slot] = b;
    }
  }
}

// ---------------------------------------------------------------------------
// Sparse fused routed-expert kernel: one block = (expert n, 32 gathered
// tokens from n's list). Only top-6 (token, expert) pairs are computed.
//   phase 1: gate/up = Xg @ Wg|Wu[:, :, n]   (WMMA f32 16x16x4, K = H)
//            h = silu(gate) * up             -> LDS  (32 x 896)
//   phase 2: out[tok] += weight * (h @ Wd[:, :, n])  (K = E, atomicAdd)
// Grid: x = expert (FAST) so all 64 experts are in flight together and every
// 128B cacheline (which interleaves 32 experts in the GGUF [.,.,N] layout) is
// consumed by all of its concurrent users while L2-resident; y = token chunk.
// A-fragments are read straight from global x (10 MB, cache-resident).
// ---------------------------------------------------------------------------
__global__ void __launch_bounds__(256, 1)
moe_routed(const float* __restrict__ x,
           const float* __restrict__ wg,   // [H, E, N]
           const float* __restrict__ wu,   // [H, E, N]
           const float* __restrict__ wd,   // [E, H, N]
           const float* __restrict__ routing,
           const int* __restrict__ counts,
           const int* __restrict__ lists,
           float* __restrict__ out) {
  __shared__ float hb[32 * ELD];   // 32 x 896 activations (padded stride)
  __shared__ float rw[32];
  __shared__ int   toks[32];

  const int n   = blockIdx.x;
  const int cnt = counts[n];
  const int t0  = blockIdx.y * 32;
  if (t0 >= cnt) return;           // nothing left on this expert's list

  const int tid  = threadIdx.x;
  const int lane = tid & 31;
  const int wave = tid >> 5;

  if (tid < 32) {
    const int valid = (t0 + tid) < cnt;
    const int tk = valid ? lists[n * BT + t0 + tid] : 0;   // dummy row 0 if pad
    toks[tid] = tk;
    rw[tid]   = valid ? routing[(size_t)tk * NEXP + n] : 0.0f;
  }
  __syncthreads();

  const int row  = lane & 15;             // A-matrix row within 16-row tile
  const int col  = lane & 15;             // B/C column
  const int koff = (lane < 16) ? 0 : 2;   // K sub-slot per lane half
  const int mb   = (lane < 16) ? 0 : 8;   // C/D row base per lane half
  const size_t dEN = (size_t)E * NEXP;    // +1 in h for gate/up weights
  const size_t dHN = (size_t)H * NEXP;    // +1 in e for down weights

  // ---- phase 1: gate/up, 2 row-tiles x 56 col-tiles ----
  for (int j = wave; j < 2 * (E / 16); j += 8) {
    const int rt  = j & 1;
    const int ec0 = (j >> 1) * 16;
    const int tk  = toks[rt * 16 + row];
    v8f ag = {0.f,0.f,0.f,0.f,0.f,0.f,0.f,0.f};
    v8f au = {0.f,0.f,0.f,0.f,0.f,0.f,0.f,0.f};
    const float* pa = x  + (size_t)tk * H + koff;
    const float* pg = wg + (size_t)n + ((size_t)koff * E + ec0 + col) * NEXP;
    const float* pu = wu + (size_t)n + ((size_t)koff * E + ec0 + col) * NEXP;
#pragma unroll 2
    for (int kk = 0; kk < H; kk += 4) {
      v2f a;  a.x  = pa[0];  a.y  = pa[1];     // coalesces to b64
      v2f bg; bg.x = pg[0];  bg.y = pg[dEN];
      v2f bu; bu.x = pu[0];  bu.y = pu[dEN];
      ag = wmma4(a, bg, ag);
      au = wmma4(a, bu, au);
      pa += 4; pg += 4 * dEN; pu += 4 * dEN;
    }
#pragma unroll
    for (int v = 0; v < 8; ++v)
      hb[(rt * 16 + mb + v) * ELD + ec0 + col] = silu_f(ag[v]) * au[v];
  }
  __syncthreads();

  // ---- phase 2: weighted down projection, 2 row-tiles x 80 col-tiles ----
  for (int j = wave; j < 2 * (H / 16); j += 8) {
    const int rt  = j & 1;
    const int hr0 = (j >> 1) * 16;
    v8f ac = {0.f,0.f,0.f,0.f,0.f,0.f,0.f,0.f};
    const float* pa = &hb[(rt * 16 + row) * ELD + koff];
    const float* pd = wd + (size_t)n + ((size_t)koff * H + hr0 + col) * NEXP;
#pragma unroll 2
    for (int e = 0; e < E; e += 4) {
      v2f a; a.x = pa[0]; a.y = pa[1];
      v2f b; b.x = pd[0]; b.y = pd[dHN];
      ac = wmma4(a, b, ac);
      pa += 4; pd += 4 * dHN;
    }
#pragma unroll
    for (int v = 0; v < 8; ++v) {
      const int m = rt * 16 + mb + v;
      const float r = rw[m];
      if (r != 0.0f)
        atomicAdd(&out[(size_t)toks[m] * H + hr0 + col], ac[v] * r);
    }
  }
}

// ---------------------------------------------------------------------------
// Shared (always-on) experts, fused the same way. FFN=1792 processed in two
// 896-column halves so the LDS activation buffer is reused. 27.5 MB of
// weights shared by all 128 blocks -> L2-resident.
// ---------------------------------------------------------------------------
__global__ void __launch_bounds__(256, 1)
moe_shared(const float* __restrict__ x,
           const float* __restrict__ wgs,   // [FFN, H]
           const float* __restrict__ wus,   // [FFN, H]
           const float* __restrict__ wds,   // [H, FFN]
           float* __restrict__ out) {
  __shared__ float xs[16 * HLD];
  __shared__ float sb[16 * ELD];   // FFN/2 == 896 columns per half

  const int tok0 = blockIdx.x * 16;
  const int tid  = threadIdx.x;
  const int lane = tid & 31;
  const int wave = tid >> 5;

  {
    const v4f* src = (const v4f*)(x + (size_t)tok0 * H);
    for (int i = tid; i < 16 * H / 4; i += 256) {
      v4f v = src[i];
      int row = (i * 4) / H, col = (i * 4) % H;
      *((v4f*)&xs[row * HLD + col]) = v;
    }
  }
  __syncthreads();

  const int row  = lane & 15;
  const int col  = lane & 15;
  const int koff = (lane < 16) ? 0 : 2;
  const int mb   = (lane < 16) ? 0 : 8;

  for (int half = 0; half < 2; ++half) {
    const int fb = half * (FFN / 2);

    // gate/up for this half of FFN; weights row-major [FFN, H] -> K contiguous
    for (int t = wave; t < (FFN / 2) / 16; t += 8) {
      const int c0 = t * 16;
      v8f ag = {0.f,0.f,0.f,0.f,0.f,0.f,0.f,0.f};
      v8f au = {0.f,0.f,0.f,0.f,0.f,0.f,0.f,0.f};
      const float* pg = wgs + (size_t)(fb + c0 + col) * H + koff;
      const float* pu = wus + (size_t)(fb + c0 + col) * H + koff;
      const float* pa = &xs[row * HLD + koff];
#pragma unroll 2
      for (int kk = 0; kk < H; kk += 4) {
        v2f a;  a.x  = pa[0]; a.y  = pa[1];
        v2f bg; bg.x = pg[0]; bg.y = pg[1];   // coalesces to b64
        v2f bu; bu.x = pu[0]; bu.y = pu[1];
        ag = wmma4(a, bg, ag);
        au = wmma4(a, bu, au);
        pa += 4; pg += 4; pu += 4;
      }
#pragma unroll
      for (int v = 0; v < 8; ++v)
        sb[(mb + v) * ELD + c0 + col] = silu_f(ag[v]) * au[v];
    }
    __syncthreads();

    // partial down projection for this half; wds [H, FFN] -> K contiguous
    for (int t = wave; t < H / 16; t += 8) {
      const int hr0 = t * 16;
      v8f ac = {0.f,0.f,0.f,0.f,0.f,0.f,0.f,0.f};
      const float* pd = wds + (size_t)(hr0 + col) * FFN + fb + koff;
      const float* pa = &sb[row * ELD + koff];
#pragma unroll 2
      for (int e = 0; e < FFN / 2; e += 4) {
        v2f a; a.x = pa[0]; a.y = pa[1];
        v2f b; b.x = pd[0]; b.y = pd[1];
        ac = wmma4(a, b, ac);
        pa += 4; pd += 4;
      }
#pragma unroll
      for (int v = 0; v < 8; ++v)
        atomicAdd(&out[(size_t)(tok0 + mb + v) * H + hr0 + col], ac[v]);
    }
    __syncthreads();   // sb reused by next half
  }
}

// ---------------------------------------------------------------------------
extern "C" void kernel_launch(void* const* d_in, const int* in_sizes, int n_in,
                              void* d_out, int out_size, void* d_ws, size_t ws_size,
                              hipStream_t stream) {
  const float* x   = (const float*)d_in[0];
  const float* wr  = (const float*)d_in[1];
  const float* wg  = (const float*)d_in[2];
  const float* wu  = (const float*)d_in[3];
  const float* wd  = (const float*)d_in[4];
  const float* wgs = (const float*)d_in[5];
  const float* wus = (const float*)d_in[6];
  const float* wds = (const float*)d_in[7];
  float* out = (float*)d_out;

  // workspace layout: routing [BT*NEXP] f32 | counts [NEXP] i32 | lists [NEXP*BT] i32
  float* routing = (float*)d_ws;
  int*   counts  = (int*)(routing + (size_t)BT * NEXP);
  int*   lists   = counts + NEXP;

  zero_out_kernel<<<(BT * H + 255) / 256, 256, 0, stream>>>(out, BT * H);
  zero_counts_kernel<<<1, NEXP, 0, stream>>>(counts);
  moe_router<<<BT, 64, 0, stream>>>(x, wr, routing, counts, lists);
  // expert = blockIdx.x (fast) so all 64 experts run concurrently; 64 token
  // chunks of 32 cover the worst case (every token routed to one expert).
  moe_routed<<<dim3(NEXP, BT / 32), 256, 0, stream>>>(
      x, wg, wu, wd, routing, counts, lists, out);
  moe_shared<<<BT / 16, 256, 0, stream>>>(x, wgs, wus, wds, out);
}